// DBASolver_19327352832526
// MI455X (gfx1250) — compile-verified
//
#include <hip/hip_runtime.h>

// ---------------------------------------------------------------------------
// DBA solver for MI455X (gfx1250, wave32).
//
// Inputs (d_in order): 0:r (B,N,2) 1:w (B,N,2) 2:J_p (B,N,2,6) 3:J_d (B,N,2,1)
// 4:lmbda (1). Output: [delta_pose (B,6) | delta_depth (B,N)] f32.
//
// Math: with u_{n,k} = [J_p[n,k,0..5], r[n,k]] (7-vec),
//   P = sum_{n,k} conf_n * u u^T        -> H_pp = P[0:6,0:6], g_p = P[0:6,6]
//   v_n = conf*(Jd0*u_{n,0} + Jd1*u_{n,1}) = [H_pd[n,0..5], g_d[n]]
//   H_dd = conf*(Jd0^2+Jd1^2) + lmbda + node_lambda; inv = 1/(H_dd+1e-6)
//   Q = sum_n inv_n * v v^T             -> Schur term + g2
//   H_eff = sym(H_pp + lmbda*I - Q[0:6,0:6]) + 1e-3*I
//   g_eff = g_p - Q[0:6,6];  dp = solve(H_eff, g_eff)
//   delta_depth[n] = inv_n * (g_d[n] - h_n . dp)
//
// P,Q accumulated with V_WMMA_F32_16X16X4_F32 (A 16x4: lane L<16 holds
// A[L][K0],A[L][K1]; lane L>=16 holds A[L-16][K2],A[L-16][K3]; B mirrored).
// P: 4 K-slots = 2 n's x 2 residual-k's per WMMA.
// Q: 4 K-slots = 4 n's per WMMA (2x-unrolled loop; shfl_xor(16) exchange).
// ---------------------------------------------------------------------------

typedef __attribute__((ext_vector_type(2))) float v2f;
typedef __attribute__((ext_vector_type(8))) float v8f;

#define DBA_B 32
#define DBA_N 65536
#define DBA_G 32                 // partial-chunks per batch (ws: 32*32*128*4 = 512 KB)
#define DBA_SPAN (DBA_N / DBA_G) // 2048 n per workgroup
#define DBA_PW (DBA_SPAN / 8)    // 256 n per wave

__device__ __forceinline__ float fast_rcp(float x) {
  float inv = __builtin_amdgcn_rcpf(x);      // v_rcp_f32 (~1 ulp after NR)
  inv = inv * (2.0f - x * inv);              // one Newton-Raphson step
  return inv;
}

__global__ __launch_bounds__(256) void dba_stage1_reduce(
    const float* __restrict__ r, const float* __restrict__ w,
    const float* __restrict__ Jp, const float* __restrict__ Jd,
    const float* __restrict__ lmbda, float* __restrict__ partial)
{
  const int b    = blockIdx.x / DBA_G;
  const int c    = blockIdx.x % DBA_G;
  const int wave = threadIdx.x >> 5;
  const int lane = threadIdx.x & 31;
  const int half = lane >> 4;     // which K-slot pair this lane feeds
  const int i    = lane & 15;     // matrix row/col component index
  const float lmb = lmbda[0];

  const int base = c * DBA_SPAN + wave * DBA_PW;
  const size_t rb0 = (size_t)b * DBA_N + base + half;  // this lane's first n

  // Loop-invariant per-lane source selection (branchless inner loop):
  //   i<6  -> u components from J_p (k=0 at +0, k=1 at +6), stride 24 floats/2n
  //   i>=6 -> from r (k=0,k=1 adjacent), stride 4 floats/2n; mask 0 for i>6
  const float* pux;
  const float* puy;
  int   strideF;
  float msk;
  if (i < 6) { pux = Jp + rb0 * 12 + i; puy = pux + 6; strideF = 24; msk = 1.f; }
  else       { pux = r + rb0 * 2;       puy = pux + 1; strideF = 4;
               msk = (i == 6) ? 1.f : 0.f; }
  const float2* pw  = (const float2*)(w  + rb0 * 2);   // (conf, node_lambda)
  const float2* pjd = (const float2*)(Jd + rb0 * 2);   // (Jd0, Jd1)

  v8f cP = {0.f,0.f,0.f,0.f,0.f,0.f,0.f,0.f};
  v8f cQ = {0.f,0.f,0.f,0.f,0.f,0.f,0.f,0.f};

  for (int t = 0; t < DBA_PW / 2; t += 2) {
    float vA, svA, vB, svB;
    // ---- sub-iteration A (n pair 2t, 2t+1) ----
    {
      // Issue all loads into independent locals first, then advance pointers,
      // then compute: keeps multiple loads in flight (no loadcnt-0 waits
      // between dependent load/use pairs).
      const float  lux = pux[0];
      const float  luy = puy[0];
      const float2 wv  = pw[0];
      const float2 jd  = pjd[0];
      pux += strideF; puy += strideF; pw += 2; pjd += 2;

      const float ux = lux * msk;
      const float uy = luy * msk;
      v2f pa, pb;
      pa.x = wv.x * ux; pa.y = wv.x * uy;
      pb.x = ux;        pb.y = uy;
      cP = __builtin_amdgcn_wmma_f32_16x16x4_f32(false, pa, false, pb,
                                                 (short)0, cP, false, false);
      vA = jd.x * pa.x + jd.y * pa.y;                 // v[i] for this lane's n
      const float hdd = wv.x * (jd.x * jd.x + jd.y * jd.y) + lmb + wv.y;
      svA = vA * fast_rcp(hdd + 1e-6f);
    }
    // ---- sub-iteration B (n pair 2t+2, 2t+3) ----
    {
      const float  lux = pux[0];
      const float  luy = puy[0];
      const float2 wv  = pw[0];
      const float2 jd  = pjd[0];
      pux += strideF; puy += strideF; pw += 2; pjd += 2;

      const float ux = lux * msk;
      const float uy = luy * msk;
      v2f pa, pb;
      pa.x = wv.x * ux; pa.y = wv.x * uy;
      pb.x = ux;        pb.y = uy;
      cP = __builtin_amdgcn_wmma_f32_16x16x4_f32(false, pa, false, pb,
                                                 (short)0, cP, false, false);
      vB = jd.x * pa.x + jd.y * pa.y;
      const float hdd = wv.x * (jd.x * jd.x + jd.y * jd.y) + lmb + wv.y;
      svB = vB * fast_rcp(hdd + 1e-6f);
    }
    // ---- Q: 4 n's in one WMMA. Slots 0,1 <- A's n0,n1 (fed by lanes<16);
    //      slots 2,3 <- B's n0,n1 (fed by lanes>=16). shfl_xor(16) swaps halves.
    const float vAo  = __shfl_xor(vA, 16, 32);
    const float svAo = __shfl_xor(svA, 16, 32);
    const float vBo  = __shfl_xor(vB, 16, 32);
    const float svBo = __shfl_xor(svB, 16, 32);
    const bool lo = (half == 0);
    v2f qa, qb;
    qa.x = lo ? svA  : svBo;  qa.y = lo ? svAo : svB;
    qb.x = lo ? vA   : vBo;   qb.y = lo ? vAo  : vB;
    cQ = __builtin_amdgcn_wmma_f32_16x16x4_f32(false, qa, false, qb,
                                               (short)0, cQ, false, false);
  }

  // C layout: rows 0..7 in vgprs 0..7 of lanes 0..15 (lane = column).
  // Useful block is rows 0..6 x cols 0..6.
  __shared__ float sred[8][112];
  if (lane < 7) {
#pragma unroll
    for (int p = 0; p < 7; ++p) {
      sred[wave][p * 7 + lane]      = cP[p];
      sred[wave][49 + p * 7 + lane] = cQ[p];
    }
  }
  __syncthreads();
  if (threadIdx.x < 98) {
    float s = 0.f;
#pragma unroll
    for (int wv = 0; wv < 8; ++wv) s += sred[wv][threadIdx.x];
    partial[(size_t)blockIdx.x * 128 + threadIdx.x] = s;  // deterministic partial
  }
}

// One block per batch: reduce partials, build H_eff/g_eff, 6x6 LU solve.
__global__ __launch_bounds__(128) void dba_stage2_solve(
    const float* __restrict__ partial, const float* __restrict__ lmbda,
    float* __restrict__ out)
{
  const int b = blockIdx.x;
  __shared__ float acc[98];
  if (threadIdx.x < 98) {
    float s = 0.f;
    for (int g = 0; g < DBA_G; ++g)
      s += partial[(size_t)(b * DBA_G + g) * 128 + threadIdx.x];
    acc[threadIdx.x] = s;
  }
  __syncthreads();
  if (threadIdx.x == 0) {
    const float lmb = lmbda[0];
    float A[6][7];
    for (int ii = 0; ii < 6; ++ii) {
      for (int jj = 0; jj < 6; ++jj) {
        const float hij = acc[ii * 7 + jj] - acc[49 + ii * 7 + jj];
        const float hji = acc[jj * 7 + ii] - acc[49 + jj * 7 + ii];
        A[ii][jj] = 0.5f * (hij + hji);
      }
      A[ii][ii] += lmb + 0.001f;
      A[ii][6] = acc[ii * 7 + 6] - acc[49 + ii * 7 + 6];
    }
    // Gaussian elimination with partial pivoting on augmented [A|g]
    for (int col = 0; col < 6; ++col) {
      int piv = col; float best = fabsf(A[col][col]);
      for (int rr = col + 1; rr < 6; ++rr) {
        const float av = fabsf(A[rr][col]);
        if (av > best) { best = av; piv = rr; }
      }
      if (piv != col)
        for (int jj = col; jj < 7; ++jj) {
          const float tmp = A[col][jj]; A[col][jj] = A[piv][jj]; A[piv][jj] = tmp;
        }
      const float invd = 1.0f / A[col][col];
      for (int rr = col + 1; rr < 6; ++rr) {
        const float f = A[rr][col] * invd;
        for (int jj = col; jj < 7; ++jj) A[rr][jj] -= f * A[col][jj];
      }
    }
    float x[6];
    for (int ii = 5; ii >= 0; --ii) {
      float s = A[ii][6];
      for (int jj = ii + 1; jj < 6; ++jj) s -= A[ii][jj] * x[jj];
      x[ii] = s / A[ii][ii];
    }
    for (int ii = 0; ii < 6; ++ii) out[b * 6 + ii] = x[ii];
  }
}

// Epilogue: delta_depth[b,n] = inv * (g_d - h . dp). One thread per (b,n).
__global__ __launch_bounds__(256) void dba_stage3_depth(
    const float* __restrict__ r, const float* __restrict__ w,
    const float* __restrict__ Jp, const float* __restrict__ Jd,
    const float* __restrict__ lmbda, const float* __restrict__ dp,
    float* __restrict__ dd)
{
  const int idx = blockIdx.x * blockDim.x + threadIdx.x;
  if (idx >= DBA_B * DBA_N) return;
  const int b = idx / DBA_N;
  const size_t rb = (size_t)idx;

  const float4* Jp4 = (const float4*)(Jp + rb * 12);  // 48B-aligned
  const float4 a0 = Jp4[0], a1 = Jp4[1], a2 = Jp4[2];
  const float2 wv = ((const float2*)w)[rb];   // (conf, node_lambda)
  const float2 jd = ((const float2*)Jd)[rb];  // (Jd0, Jd1)
  const float2 rv = ((const float2*)r)[rb];   // (r0, r1)
  const float lmb = lmbda[0];

  const float p0 = dp[b * 6 + 0], p1 = dp[b * 6 + 1], p2 = dp[b * 6 + 2];
  const float p3 = dp[b * 6 + 3], p4 = dp[b * 6 + 4], p5 = dp[b * 6 + 5];

  // h_i = conf*(jd0*Jp[k=0][i] + jd1*Jp[k=1][i])
  const float h0 = wv.x * (jd.x * a0.x + jd.y * a1.z);
  const float h1 = wv.x * (jd.x * a0.y + jd.y * a1.w);
  const float h2 = wv.x * (jd.x * a0.z + jd.y * a2.x);
  const float h3 = wv.x * (jd.x * a0.w + jd.y * a2.y);
  const float h4 = wv.x * (jd.x * a1.x + jd.y * a2.z);
  const float h5 = wv.x * (jd.x * a1.y + jd.y * a2.w);
  const float gd  = wv.x * (jd.x * rv.x + jd.y * rv.y);
  const float hdd = wv.x * (jd.x * jd.x + jd.y * jd.y) + lmb + wv.y;
  const float inv = fast_rcp(hdd + 1e-6f);
  const float vdot = h0 * p0 + h1 * p1 + h2 * p2 + h3 * p3 + h4 * p4 + h5 * p5;
  dd[idx] = inv * (gd - vdot);
}

extern "C" void kernel_launch(void* const* d_in, const int* in_sizes, int n_in,
                              void* d_out, int out_size, void* d_ws, size_t ws_size,
                              hipStream_t stream) {
  const float* r   = (const float*)d_in[0];
  const float* w   = (const float*)d_in[1];
  const float* Jp  = (const float*)d_in[2];
  const float* Jd  = (const float*)d_in[3];
  const float* lmb = (const float*)d_in[4];
  float* out = (float*)d_out;              // [dp (32*6) | dd (32*65536)]
  float* ws  = (float*)d_ws;               // needs 32*32*128*4 = 512 KB

  dba_stage1_reduce<<<dim3(DBA_B * DBA_G), dim3(256), 0, stream>>>(
      r, w, Jp, Jd, lmb, ws);
  dba_stage2_solve<<<dim3(DBA_B), dim3(128), 0, stream>>>(ws, lmb, out);
  const int tot = DBA_B * DBA_N;
  dba_stage3_depth<<<dim3((tot + 255) / 256), dim3(256), 0, stream>>>(
      r, w, Jp, Jd, lmb, out, out + DBA_B * 6);
}